// GCN_2602750181462
// MI455X (gfx1250) — compile-verified
//
#include <hip/hip_runtime.h>

#define NNODES 50000
#define EDGES  800000
#define EPSBN  1e-5f

typedef float v2f __attribute__((ext_vector_type(2)));
typedef float v8f __attribute__((ext_vector_type(8)));

// Guaranteed-native f32 atomic add (no return -> STOREcnt-tracked;
// S_ENDPGM's implicit wait-idle covers completion at kernel end).
__device__ __forceinline__ void atomic_add_f32(float* p, float v) {
    asm volatile("global_atomic_add_f32 %0, %1, off" :: "v"(p), "v"(v) : "memory");
}

// ---------------------------------------------------------------- zero fill
__global__ void zero_kernel(float* __restrict__ p, long n) {
    long i = (long)blockIdx.x * blockDim.x + threadIdx.x;
    long stride = (long)gridDim.x * blockDim.x;
    for (; i < n; i += stride) p[i] = 0.0f;
}

// ------------------------------------------- edge gather * w -> scatter-add
// One wave (32 lanes) per edge; float4 row chunks; agg[dst] += x[src] * ew.
// agg (51 MB) is L2-resident: L2 atomic throughput bounds this phase.
__global__ void scatter_kernel(const float* __restrict__ x,
                               const int* __restrict__ src,
                               const int* __restrict__ dst,
                               const float* __restrict__ ew,
                               float* __restrict__ agg, int F) {
    int gtid = blockIdx.x * blockDim.x + threadIdx.x;
    int edge = gtid >> 5;
    int lane = gtid & 31;
    if (edge >= EDGES) return;
    int s = src[edge], d = dst[edge];
    float w = ew[edge];
    const float4* xr = (const float4*)(x + (long)s * F);
    float* ar = agg + (long)d * F;
    int nq = F >> 2;
    for (int q = lane; q < nq; q += 32) {
        float4 v = xr[q];
        float* a = ar + q * 4;
        atomic_add_f32(a + 0, v.x * w);
        atomic_add_f32(a + 1, v.y * w);
        atomic_add_f32(a + 2, v.z * w);
        atomic_add_f32(a + 3, v.w * w);
    }
}

// --------------------------------------------------- fused dual GEMM (WMMA)
// out[n,fo] = sum_k A1[n,k]*W1[fo,k] + sum_k A2[n,k]*W2[fo,k] + bias[fo]
// fp32 WMMA 16x16x4. One wave computes a 16x64 output strip: 4 accumulators
// share each A-fragment load (4x arithmetic intensity, 4 independent WMMA
// chains to hide latency). Both K-loops (rel and root term) share the accs.
__global__ void gemm_dual_wmma(const float* __restrict__ A1,
                               const float* __restrict__ W1, int K1,
                               const float* __restrict__ A2,
                               const float* __restrict__ W2, int K2,
                               const float* __restrict__ bias,
                               float* __restrict__ out, int Nrows, int Fout) {
    int wavesPerBlock = blockDim.x >> 5;
    int wave = blockIdx.x * wavesPerBlock + (threadIdx.x >> 5);
    int lane = threadIdx.x & 31;
    int ntiles = Fout >> 6;              // 64-wide strips
    int mtiles = Nrows >> 4;
    if (wave >= ntiles * mtiles) return; // wave-uniform: EXEC stays all-ones
    int tm = wave / ntiles;
    int tn = wave % ntiles;
    long row0 = (long)tm * 16;
    int col0 = tn * 64;

    int lrow  = lane & 15;          // m for A-tile, n for B-tile
    int khalf = (lane >> 4) << 1;   // K sub-offset: 0 or 2

    v8f acc0 = {}, acc1 = {}, acc2 = {}, acc3 = {};
    {
        const float* a  = A1 + (row0 + lrow) * K1 + khalf;
        const float* b0 = W1 + (long)(col0 + 0  + lrow) * K1 + khalf;
        const float* b1 = W1 + (long)(col0 + 16 + lrow) * K1 + khalf;
        const float* b2 = W1 + (long)(col0 + 32 + lrow) * K1 + khalf;
        const float* b3 = W1 + (long)(col0 + 48 + lrow) * K1 + khalf;
#pragma unroll 4
        for (int k = 0; k < K1; k += 4) {
            v2f av = *(const v2f*)(a + k);
            v2f bv0 = *(const v2f*)(b0 + k);
            v2f bv1 = *(const v2f*)(b1 + k);
            v2f bv2 = *(const v2f*)(b2 + k);
            v2f bv3 = *(const v2f*)(b3 + k);
            acc0 = __builtin_amdgcn_wmma_f32_16x16x4_f32(false, av, false, bv0, (short)0, acc0, false, false);
            acc1 = __builtin_amdgcn_wmma_f32_16x16x4_f32(false, av, false, bv1, (short)0, acc1, false, false);
            acc2 = __builtin_amdgcn_wmma_f32_16x16x4_f32(false, av, false, bv2, (short)0, acc2, false, false);
            acc3 = __builtin_amdgcn_wmma_f32_16x16x4_f32(false, av, false, bv3, (short)0, acc3, false, false);
        }
    }
    {
        const float* a  = A2 + (row0 + lrow) * K2 + khalf;
        const float* b0 = W2 + (long)(col0 + 0  + lrow) * K2 + khalf;
        const float* b1 = W2 + (long)(col0 + 16 + lrow) * K2 + khalf;
        const float* b2 = W2 + (long)(col0 + 32 + lrow) * K2 + khalf;
        const float* b3 = W2 + (long)(col0 + 48 + lrow) * K2 + khalf;
#pragma unroll 4
        for (int k = 0; k < K2; k += 4) {
            v2f av = *(const v2f*)(a + k);
            v2f bv0 = *(const v2f*)(b0 + k);
            v2f bv1 = *(const v2f*)(b1 + k);
            v2f bv2 = *(const v2f*)(b2 + k);
            v2f bv3 = *(const v2f*)(b3 + k);
            acc0 = __builtin_amdgcn_wmma_f32_16x16x4_f32(false, av, false, bv0, (short)0, acc0, false, false);
            acc1 = __builtin_amdgcn_wmma_f32_16x16x4_f32(false, av, false, bv1, (short)0, acc1, false, false);
            acc2 = __builtin_amdgcn_wmma_f32_16x16x4_f32(false, av, false, bv2, (short)0, acc2, false, false);
            acc3 = __builtin_amdgcn_wmma_f32_16x16x4_f32(false, av, false, bv3, (short)0, acc3, false, false);
        }
    }
    // C/D layout: VGPR i -> row = i + 8*(lane>>4), col = lane&15
    long rbase = row0 + ((lane >> 4) << 3);
    v8f accs[4] = {acc0, acc1, acc2, acc3};
#pragma unroll
    for (int j = 0; j < 4; ++j) {
        int ocol = col0 + j * 16 + lrow;
        float bb = bias[ocol];
#pragma unroll
        for (int i = 0; i < 8; ++i)
            out[(rbase + i) * Fout + ocol] = accs[j][i] + bb;
    }
}

// ------------------------------------------------- batchnorm column stats
// blockDim.x == F; each thread owns one feature column -> coalesced rows.
__global__ void bn_stats_kernel(const float* __restrict__ h,
                                float* __restrict__ stats, int Nrows, int F) {
    int f = threadIdx.x;
    float s = 0.0f, s2 = 0.0f;
    for (int r = blockIdx.x; r < Nrows; r += gridDim.x) {
        float v = h[(long)r * F + f];
        s += v;
        s2 += v * v;
    }
    atomic_add_f32(&stats[f], s);
    atomic_add_f32(&stats[F + f], s2);
}

__global__ void bn_apply_kernel(float* __restrict__ h,
                                const float* __restrict__ stats,
                                const float* __restrict__ gamma,
                                const float* __restrict__ beta,
                                long n, int F) {
    long i = (long)blockIdx.x * blockDim.x + threadIdx.x;
    long stride = (long)gridDim.x * blockDim.x;
    const float invN = 1.0f / (float)NNODES;
    for (; i < n; i += stride) {
        int f = (int)(i % F);
        float mu = stats[f] * invN;
        float var = stats[F + f] * invN - mu * mu;
        float v = (h[i] - mu) * rsqrtf(var + EPSBN) * gamma[f] + beta[f];
        h[i] = v > 0.0f ? v : 0.0f;
    }
}

// ---------------------------------------------------------------------------
extern "C" void kernel_launch(void* const* d_in, const int* in_sizes, int n_in,
                              void* d_out, int out_size, void* d_ws, size_t ws_size,
                              hipStream_t stream) {
    const float* x   = (const float*)d_in[0];
    const int*   ei  = (const int*)d_in[1];
    const float* ew  = (const float*)d_in[2];
    const float* Wr0 = (const float*)d_in[3];
    const float* br0 = (const float*)d_in[4];
    const float* Wt0 = (const float*)d_in[5];
    const float* g0  = (const float*)d_in[6];
    const float* be0 = (const float*)d_in[7];
    const float* Wr1 = (const float*)d_in[8];
    const float* br1 = (const float*)d_in[9];
    const float* Wt1 = (const float*)d_in[10];
    const float* g1  = (const float*)d_in[11];
    const float* be1 = (const float*)d_in[12];
    const float* Wr2 = (const float*)d_in[13];
    const float* br2 = (const float*)d_in[14];
    const float* Wt2 = (const float*)d_in[15];
    float* out = (float*)d_out;

    const int* srcv = ei;
    const int* dstv = ei + EDGES;

    float* agg   = (float*)d_ws;                    // [N,256] max
    float* hA    = agg + (long)NNODES * 256;        // [N,256]
    float* hB    = hA  + (long)NNODES * 256;        // [N,256]
    float* stats = hB  + (long)NNODES * 256;        // [2,256]

    const int B = 256;
    const int zeroBlocks    = 2048;
    const int scatterBlocks = (EDGES * 32 + B - 1) / B;
    const int bnBlocks      = 512;

    auto gemmBlocks = [](int mtiles, int ntiles64) {
        return (mtiles * ntiles64 + 7) / 8;   // 8 waves per 256-thread block
    };

    // ---------------- Layer 0: 128 -> 256 ----------------
    zero_kernel<<<zeroBlocks, B, 0, stream>>>(agg, (long)NNODES * 128);
    scatter_kernel<<<scatterBlocks, B, 0, stream>>>(x, srcv, dstv, ew, agg, 128);
    gemm_dual_wmma<<<gemmBlocks(NNODES / 16, 256 / 64), B, 0, stream>>>(
        agg, Wr0, 128, x, Wt0, 128, br0, hA, NNODES, 256);
    zero_kernel<<<1, B, 0, stream>>>(stats, 512);
    bn_stats_kernel<<<bnBlocks, 256, 0, stream>>>(hA, stats, NNODES, 256);
    bn_apply_kernel<<<zeroBlocks, B, 0, stream>>>(hA, stats, g0, be0,
                                                  (long)NNODES * 256, 256);

    // ---------------- Layer 1: 256 -> 256 ----------------
    zero_kernel<<<zeroBlocks, B, 0, stream>>>(agg, (long)NNODES * 256);
    scatter_kernel<<<scatterBlocks, B, 0, stream>>>(hA, srcv, dstv, ew, agg, 256);
    gemm_dual_wmma<<<gemmBlocks(NNODES / 16, 256 / 64), B, 0, stream>>>(
        agg, Wr1, 256, hA, Wt1, 256, br1, hB, NNODES, 256);
    zero_kernel<<<1, B, 0, stream>>>(stats, 512);
    bn_stats_kernel<<<bnBlocks, 256, 0, stream>>>(hB, stats, NNODES, 256);
    bn_apply_kernel<<<zeroBlocks, B, 0, stream>>>(hB, stats, g1, be1,
                                                  (long)NNODES * 256, 256);

    // ---------------- Layer 2: 256 -> 64 -----------------
    zero_kernel<<<zeroBlocks, B, 0, stream>>>(agg, (long)NNODES * 256);
    scatter_kernel<<<scatterBlocks, B, 0, stream>>>(hB, srcv, dstv, ew, agg, 256);
    gemm_dual_wmma<<<gemmBlocks(NNODES / 16, 64 / 64), B, 0, stream>>>(
        agg, Wr2, 256, hB, Wt2, 256, br2, out, NNODES, 64);
}